// Conv2D_6605659701961
// MI455X (gfx1250) — compile-verified
//
#include <hip/hip_runtime.h>

// CDNA5 (gfx1250) fp32 implicit-GEMM 3x3 conv using V_WMMA_F32_16X16X4_F32.
// Block tile: 32 C_out x 8 rows x 64 cols (56 valid). One row per wave32,
// each wave computes 2 (M) x 4 (N) 16x16 accumulators = 8 WMMA per K-step.
// Both operands staged in LDS; x staging uses GLOBAL_LOAD_ASYNC_TO_LDS_B128
// (ASYNCcnt) when the builtin exists, else VGPR staging.

typedef __attribute__((ext_vector_type(2))) float v2f;
typedef __attribute__((ext_vector_type(8))) float v8f;
typedef __attribute__((ext_vector_type(4))) int   v4i;

#define B_SZ   16
#define C_IN   128
#define H_SZ   56
#define W_SZ   56
#define HW     (H_SZ * W_SZ)          // 3136
#define C_OUT  256
#define K_TOT  (C_IN * 9)             // 1152

#define CI_CHUNK   32                 // input channels staged per pass
#define MT         2                  // 2 x 16 = 32 output channels per wave
#define NT         4                  // 4 x 16 = 64 output cols per wave (56 valid)
#define WAVES      8                  // one output row per wave
#define BLOCK      (WAVES * 32)       // 256 threads

// x tile: CI_CHUNK planes x 10 rows (8+halo) x 72 cols. Valid data starts at
// col 4 (so global/LDS 16B alignment match for B128 async copies); left halo
// col 3, right halo col 60. 2*720 % 64 == 32 -> half-wave K-split is bank-clean.
#define XROW   72
#define XPLANE (10 * XROW)            // 720
#define XSZ    (CI_CHUNK * XPLANE)    // 23040 floats = 90*256 (90 KB)
#define XCOL0  4                      // LDS col of w_in = 0
// A tile: [ci][tap][m], stride 33 (coprime with 64 banks): (c*9+k)*33 + m
#define ASTR   33
#define ASZ    (CI_CHUNK * 9 * ASTR)  // 9504 floats (37.1 KB)

#if __has_builtin(__builtin_amdgcn_global_load_async_to_lds_b128)
#define USE_ASYNC_LDS 1
#endif

__global__ __launch_bounds__(BLOCK)
void conv3x3_wmma_f32(const float* __restrict__ x,
                      const float* __restrict__ Wt,
                      const float* __restrict__ bias,
                      float* __restrict__ out)
{
    __shared__ float xs[XSZ];
    __shared__ float as[ASZ];

    const int tid  = threadIdx.x;
    const int wave = tid >> 5;
    const int lane = tid & 31;
    const int l15  = lane & 15;
    const int koff = (lane < 16) ? 0 : 2;   // K-pair of this half-wave

    const int co0 = blockIdx.x * (MT * 16); // C_out tile (32 wide)
    const int h0  = blockIdx.y * WAVES;     // row group (7 x 8 = 56)
    const int bb  = blockIdx.z;             // batch
    const int h   = h0 + wave;              // this wave's output row

    v8f acc[MT][NT];
#pragma unroll
    for (int mi = 0; mi < MT; ++mi)
#pragma unroll
        for (int t = 0; t < NT; ++t)
#pragma unroll
            for (int r = 0; r < 8; ++r)
                acc[mi][t][r] = 0.0f;

#if USE_ASYNC_LDS
    // One-time zero of the whole x tile: halo cells are geometry-invariant
    // across ci-chunks, so async passes only ever rewrite the valid cells.
#pragma unroll 4
    for (int i = 0; i < XSZ / BLOCK; ++i)
        xs[i * BLOCK + tid] = 0.0f;
#endif

    for (int ci0 = 0; ci0 < C_IN; ci0 += CI_CHUNK) {
        __syncthreads();

#if USE_ASYNC_LDS
        // ---- x tile via async B128 copies: 32 planes x 10 rows x 14 x 16B.
        // Both src and dst are 16B aligned (row origin at col 4, wi % 4 == 0).
        for (int i = 0; i < 18; ++i) {
            int e = i * BLOCK + tid;            // 4480 = 17.5 * 256
            if (e < CI_CHUNK * 10 * 14) {
                int c   = e / 140;
                int rem = e - c * 140;
                int r   = rem / 14;
                int wi  = (rem - r * 14) * 4;
                int hi  = h0 + r - 1;
                if ((unsigned)hi < (unsigned)H_SZ) {
                    const float* src = x + ((bb * C_IN + ci0 + c) * H_SZ + hi) * W_SZ + wi;
                    float*       dst = xs + c * XPLANE + r * XROW + XCOL0 + wi;
                    __builtin_amdgcn_global_load_async_to_lds_b128(
                        (v4i*)src, (v4i*)dst, 0, 0);
                }
            }
        }
#else
        // ---- fallback: VGPR staging with halo zero-blend.
#pragma unroll 4
        for (int i = 0; i < XSZ / BLOCK; ++i) {
            int e   = i * BLOCK + tid;
            int c   = e / XPLANE;
            int rem = e - c * XPLANE;
            int r   = rem / XROW;
            int wi  = (rem - r * XROW) - XCOL0;
            int hi  = h0 + r - 1;
            float v = 0.0f;
            if (((unsigned)hi < (unsigned)H_SZ) && ((unsigned)wi < (unsigned)W_SZ))
                v = x[((bb * C_IN + ci0 + c) * H_SZ + hi) * W_SZ + wi];
            xs[e] = v;
        }
#endif

        // ---- A tile: W[co0..+31][ci0..+31][0..8]; contiguous 288-float runs
        // per co (coalesced); LDS stride 33 -> conflict-free strided stores.
        // Issued while the async x copies are still in flight.
#pragma unroll
        for (int i = 0; i < (MT * 16 * CI_CHUNK * 9) / BLOCK; ++i) {
            int e = i * BLOCK + tid;
            int m = e / (CI_CHUNK * 9);         // 0..31
            int r = e - m * (CI_CHUNK * 9);     // c*9 + k
            as[r * ASTR + m] = Wt[(co0 + m) * K_TOT + ci0 * 9 + r];
        }

#if USE_ASYNC_LDS
        // Drain this wave's async copies only now, right before the barrier.
#if __has_builtin(__builtin_amdgcn_s_wait_asynccnt)
        __builtin_amdgcn_s_wait_asynccnt(0);
#else
        asm volatile("s_wait_asynccnt 0x0" ::: "memory");
#endif
#endif
        __syncthreads();

        for (int kidx = 0; kidx < 9; ++kidx) {
            const int kh = kidx / 3;
            const int kw = kidx - kh * 3;
            int bbase = koff * XPLANE + (wave + kh) * XROW + (kw - 1) + XCOL0 + l15;
            int abase = (koff * 9 + kidx) * ASTR + l15;

#pragma unroll
            for (int c4 = 0; c4 < CI_CHUNK; c4 += 4) {
                v2f a0, a1;
                a0.x = as[abase];
                a0.y = as[abase + 9 * ASTR];        // ci+1
                a1.x = as[abase + 16];              // second M-subtile
                a1.y = as[abase + 9 * ASTR + 16];
                v2f bf[NT];
#pragma unroll
                for (int t = 0; t < NT; ++t) {
                    bf[t].x = xs[bbase + t * 16];
                    bf[t].y = xs[bbase + t * 16 + XPLANE];
                }
#pragma unroll
                for (int t = 0; t < NT; ++t) {
                    acc[0][t] = __builtin_amdgcn_wmma_f32_16x16x4_f32(
                        false, a0, false, bf[t], (short)0, acc[0][t], false, false);
                    acc[1][t] = __builtin_amdgcn_wmma_f32_16x16x4_f32(
                        false, a1, false, bf[t], (short)0, acc[1][t], false, false);
                }
                abase += 4 * 9 * ASTR;
                bbase += 4 * XPLANE;
            }
        }
    }

    // Epilogue: D layout — VGPR r: M = r + (lane<16 ? 0 : 8), N(col) = t*16 + l15
    const int mh = (lane < 16) ? 0 : 8;
#pragma unroll
    for (int t = 0; t < NT; ++t) {
        int w = t * 16 + l15;
        if (w < W_SZ) {
#pragma unroll
            for (int mi = 0; mi < MT; ++mi)
#pragma unroll
                for (int r = 0; r < 8; ++r) {
                    int co = co0 + mi * 16 + r + mh;
                    float v = acc[mi][t][r] + bias[co];
                    __builtin_nontemporal_store(
                        v, &out[((bb * C_OUT + co) * H_SZ + h) * W_SZ + w]);
                }
        }
    }
}

extern "C" void kernel_launch(void* const* d_in, const int* in_sizes, int n_in,
                              void* d_out, int out_size, void* d_ws, size_t ws_size,
                              hipStream_t stream) {
    (void)in_sizes; (void)n_in; (void)d_ws; (void)ws_size; (void)out_size;
    const float* x    = (const float*)d_in[0];   // (16,128,56,56)
    const float* Wt   = (const float*)d_in[1];   // (256,128,3,3)
    const float* bias = (const float*)d_in[2];   // (256,)
    float*       out  = (float*)d_out;           // (16,256,56,56)

    dim3 grid(C_OUT / (MT * 16), H_SZ / WAVES, B_SZ);   // (8, 7, 16)
    conv3x3_wmma_f32<<<grid, BLOCK, 0, stream>>>(x, Wt, bias, out);
}